// SupportGAT_386547057271
// MI455X (gfx1250) — compile-verified
//
#include <hip/hip_runtime.h>
#include <math.h>

typedef float v2f __attribute__((ext_vector_type(2)));
typedef float v8f __attribute__((ext_vector_type(8)));

#define NEG_SLOPE 0.2f

// ---- order-preserving float <-> uint encoding (for atomic segment max) ----
__device__ __forceinline__ unsigned fenc(float f) {
  unsigned u = __float_as_uint(f);
  return (u & 0x80000000u) ? ~u : (u | 0x80000000u);
}
__device__ __forceinline__ float fdec(unsigned e) {
  return (e & 0x80000000u) ? __uint_as_float(e ^ 0x80000000u)
                           : __uint_as_float(~e);
}
// enc(-inf) = ~0xFF800000 = 0x007FFFFF
#define ENC_NEG_INF 0x007FFFFFu

// ---------------------------------------------------------------------------
// generic fill (zeros / -inf codes)
__global__ void fill_u32(unsigned* __restrict__ p, unsigned v, long long n) {
  long long i = (long long)blockIdx.x * blockDim.x + threadIdx.x;
  if (i < n) p[i] = v;
}

// ---------------------------------------------------------------------------
// fp32 WMMA GEMM: C[M,N] = A[M,K] @ B[K,N], row-major.
// One wave computes a 16x64 tile (4x v_wmma_f32_16x16x4_f32 per K-step).
// Requires M%16==0, N%64==0, K%4==0 (true for all shapes here).
// Optional epilogue: +bias[col], relu.
__global__ void gemm_wmma_f32(const float* __restrict__ A,
                              const float* __restrict__ B,
                              float* __restrict__ C,
                              int M, int N, int K,
                              const float* __restrict__ bias, int relu) {
  int wavesPerBlock = blockDim.x >> 5;
  int wave = blockIdx.x * wavesPerBlock + (threadIdx.x >> 5);
  int lane = threadIdx.x & 31;
  int ng = N >> 6;                 // 64-wide column groups
  int tm = wave / ng;
  int tg = wave - tm * ng;
  int m0 = tm << 4;
  if (m0 >= M) return;
  int n0 = tg << 6;
  int row = lane & 15;             // A row within tile / B-D column within 16
  int kh  = (lane >> 4) << 1;      // lanes 0-15: K=0,1 ; lanes 16-31: K=2,3

  v8f c0 = {}, c1 = {}, c2 = {}, c3 = {};
  const float* Ar = A + (size_t)(m0 + row) * K + kh;

  for (int k = 0; k < K; k += 4) {
    v2f a;
    a.x = Ar[k];
    a.y = Ar[k + 1];
    const float* B0 = B + (size_t)(k + kh) * N + n0 + row;
    const float* B1 = B0 + N;
    v2f b0, b1, b2, b3;
    b0.x = B0[0];  b0.y = B1[0];
    b1.x = B0[16]; b1.y = B1[16];
    b2.x = B0[32]; b2.y = B1[32];
    b3.x = B0[48]; b3.y = B1[48];
    c0 = __builtin_amdgcn_wmma_f32_16x16x4_f32(false, a, false, b0, (short)0, c0, false, false);
    c1 = __builtin_amdgcn_wmma_f32_16x16x4_f32(false, a, false, b1, (short)0, c1, false, false);
    c2 = __builtin_amdgcn_wmma_f32_16x16x4_f32(false, a, false, b2, (short)0, c2, false, false);
    c3 = __builtin_amdgcn_wmma_f32_16x16x4_f32(false, a, false, b3, (short)0, c3, false, false);
  }

  int rb = m0 + ((lane >> 4) << 3);   // lanes 0-15 -> rows v, lanes 16-31 -> rows v+8
  for (int v = 0; v < 8; ++v) {
    float o0 = c0[v], o1 = c1[v], o2 = c2[v], o3 = c3[v];
    if (bias) {
      o0 += bias[n0 + row];
      o1 += bias[n0 + row + 16];
      o2 += bias[n0 + row + 32];
      o3 += bias[n0 + row + 48];
    }
    if (relu) {
      o0 = fmaxf(o0, 0.f); o1 = fmaxf(o1, 0.f);
      o2 = fmaxf(o2, 0.f); o3 = fmaxf(o3, 0.f);
    }
    float* Cr = C + (size_t)(rb + v) * N + n0 + row;
    Cr[0]  = o0;
    Cr[16] = o1;
    Cr[32] = o2;
    Cr[48] = o3;
  }
}

// ---------------------------------------------------------------------------
// per-node attention coefficients: aS[n,h] = H[n,h,:]·attS[h,:], same for aD.
// one wave per node; wave32 shuffle reduction.
__global__ void attn_coef(const float* __restrict__ H,
                          const float* __restrict__ attS,
                          const float* __restrict__ attD,
                          float* __restrict__ aS, float* __restrict__ aD,
                          int Nn, int heads, int C) {
  int wave = blockIdx.x * (blockDim.x >> 5) + (threadIdx.x >> 5);
  int lane = threadIdx.x & 31;
  if (wave >= Nn) return;
  for (int h = 0; h < heads; ++h) {
    const float* Hp = H + ((size_t)wave * heads + h) * C;
    const float* as = attS + h * C;
    const float* ad = attD + h * C;
    float s = 0.f, d = 0.f;
    for (int c = lane; c < C; c += 32) {
      float v = Hp[c];
      s += v * as[c];
      d += v * ad[c];
    }
    for (int off = 16; off; off >>= 1) {
      s += __shfl_xor(s, off, 32);
      d += __shfl_xor(d, off, 32);
    }
    if (lane == 0) {
      aS[wave * heads + h] = s;
      aD[wave * heads + h] = d;
    }
  }
}

__device__ __forceinline__ void edge_sd(int e, int E,
                                        const int* __restrict__ srcI,
                                        const int* __restrict__ dstI,
                                        int& s, int& d) {
  if (e < E) { s = srcI[e]; d = dstI[e]; }
  else       { s = e - E;   d = e - E; }   // self loops appended
}

// pass 1: segment max of leaky_relu(aS[src]+aD[dst]) per (dst, head)
__global__ void edge_max_k(const int* __restrict__ srcI, const int* __restrict__ dstI,
                           const float* __restrict__ aS, const float* __restrict__ aD,
                           unsigned* __restrict__ emax, int E, int ET, int heads) {
  int t = blockIdx.x * blockDim.x + threadIdx.x;
  if (t >= ET * heads) return;
  int e = t / heads, h = t - e * heads;
  int s, d; edge_sd(e, E, srcI, dstI, s, d);
  float v = aS[s * heads + h] + aD[d * heads + h];
  v = (v > 0.f) ? v : NEG_SLOPE * v;
  atomicMax(&emax[d * heads + h], fenc(v));
}

// pass 2: e_exp and segment sum of exp
__global__ void edge_expsum_k(const int* __restrict__ srcI, const int* __restrict__ dstI,
                              const float* __restrict__ aS, const float* __restrict__ aD,
                              const unsigned* __restrict__ emax,
                              float* __restrict__ eexp, float* __restrict__ den,
                              int E, int ET, int heads) {
  int t = blockIdx.x * blockDim.x + threadIdx.x;
  if (t >= ET * heads) return;
  int e = t / heads, h = t - e * heads;
  int s, d; edge_sd(e, E, srcI, dstI, s, d);
  float v = aS[s * heads + h] + aD[d * heads + h];
  v = (v > 0.f) ? v : NEG_SLOPE * v;
  float ex = __expf(v - fdec(emax[d * heads + h]));
  eexp[(size_t)e * heads + h] = ex;
  atomicAdd(&den[d * heads + h], ex);
}

// pass 3: agg[dst] += alpha * H[src]; 4 channels per thread, float4 gather
__global__ void edge_scatter_k(const int* __restrict__ srcI, const int* __restrict__ dstI,
                               const float* __restrict__ H,
                               const float* __restrict__ eexp,
                               const float* __restrict__ den,
                               float* __restrict__ agg,
                               int E, int ET, int heads, int C) {
  int HC = heads * C;
  int tpe = HC >> 2;
  long long t = (long long)blockIdx.x * blockDim.x + threadIdx.x;
  if (t >= (long long)ET * tpe) return;
  int e = (int)(t / tpe);
  int ci = (int)(t - (long long)e * tpe) << 2;   // channel base, 4 contiguous in 1 head
  int s, d; edge_sd(e, E, srcI, dstI, s, d);
  int h = ci / C;
  float alpha = eexp[(size_t)e * heads + h] / (den[d * heads + h] + 1e-16f);
  float4 hv = *(const float4*)(H + (size_t)s * HC + ci);
  float* Ap = agg + (size_t)d * HC + ci;
  atomicAdd(Ap + 0, hv.x * alpha);
  atomicAdd(Ap + 1, hv.y * alpha);
  atomicAdd(Ap + 2, hv.z * alpha);
  atomicAdd(Ap + 3, hv.w * alpha);
}

// mean over heads + bias, then ELU
__global__ void finalize_elu(const float* __restrict__ agg,
                             const float* __restrict__ bias,
                             float* __restrict__ out, int Nn, int heads, int C) {
  int t = blockIdx.x * blockDim.x + threadIdx.x;
  if (t >= Nn * C) return;
  int n = t / C, c = t - n * C;
  float s = 0.f;
  for (int h = 0; h < heads; ++h) s += agg[((size_t)n * heads + h) * C + c];
  s = s / (float)heads + bias[c];
  out[(size_t)n * C + c] = (s > 0.f) ? s : (__expf(s) - 1.0f);
}

// global mean pool: accumulate
__global__ void pool_add_k(const float* __restrict__ h, const int* __restrict__ batch,
                           float* __restrict__ pooled, int Nn, int C) {
  int t = blockIdx.x * blockDim.x + threadIdx.x;
  if (t >= Nn * C) return;
  int n = t / C, c = t - n * C;
  atomicAdd(&pooled[(size_t)batch[n] * C + c], h[(size_t)n * C + c]);
}

__global__ void count_k(const int* __restrict__ batch, float* __restrict__ cnt, int Nn) {
  int t = blockIdx.x * blockDim.x + threadIdx.x;
  if (t < Nn) atomicAdd(&cnt[batch[t]], 1.0f);
}

__global__ void pool_div_k(float* __restrict__ pooled, const float* __restrict__ cnt,
                           int G, int C) {
  int t = blockIdx.x * blockDim.x + threadIdx.x;
  if (t >= G * C) return;
  int g = t / C;
  pooled[t] /= fmaxf(cnt[g], 1.0f);
}

// final linear [G,C]@[C,1] + b
__global__ void head_logits_k(const float* __restrict__ z, const float* __restrict__ Wc2,
                              const float* __restrict__ bc2, float* __restrict__ out,
                              int G, int C) {
  int g = blockIdx.x * blockDim.x + threadIdx.x;
  if (g >= G) return;
  float s = bc2[0];
  const float* zp = z + (size_t)g * C;
  for (int c = 0; c < C; ++c) s += zp[c] * Wc2[c];
  out[g] = s;
}

// ---------------------------------------------------------------------------
static inline int ceil_div_i(long long a, long long b) { return (int)((a + b - 1) / b); }

extern "C" void kernel_launch(void* const* d_in, const int* in_sizes, int n_in,
                              void* d_out, int out_size, void* d_ws, size_t ws_size,
                              hipStream_t stream) {
  const float* x     = (const float*)d_in[0];
  const int*   ei    = (const int*)d_in[1];     // [2, E] flat: src then dst
  const int*   batch = (const int*)d_in[2];
  const float* W1    = (const float*)d_in[3];
  const float* attS1 = (const float*)d_in[4];
  const float* attD1 = (const float*)d_in[5];
  const float* b1    = (const float*)d_in[6];
  const float* W2    = (const float*)d_in[7];
  const float* attS2 = (const float*)d_in[8];
  const float* attD2 = (const float*)d_in[9];
  const float* b2    = (const float*)d_in[10];
  const float* Wc1   = (const float*)d_in[11];
  const float* bc1   = (const float*)d_in[12];
  const float* Wc2   = (const float*)d_in[13];
  const float* bc2   = (const float*)d_in[14];

  const int Nn = in_sizes[2];            // 40000 nodes
  const int E  = in_sizes[1] / 2;        // 640000 edges
  const int ET = E + Nn;                 // + self loops
  const int C  = in_sizes[6];            // 128 hidden
  const int H1HEADS = in_sizes[4] / C;   // 2
  const int IN = in_sizes[0] / Nn;       // 256
  const int HC1 = H1HEADS * C;           // 256
  const int G = out_size;                // 64 graphs
  const int* srcI = ei;
  const int* dstI = ei + E;

  // -------- workspace layout (floats), with cross-layer reuse --------
  float* ws = (float*)d_ws;
  size_t o = 0;
  float*    H1    = ws + o; o += (size_t)Nn * HC1;        // layer1 features [N, 2*128]
  float*    aS    = ws + o; o += (size_t)Nn * H1HEADS;
  float*    aD    = ws + o; o += (size_t)Nn * H1HEADS;
  unsigned* emax  = (unsigned*)(ws + o); o += (size_t)Nn * H1HEADS;
  float*    den   = ws + o; o += (size_t)Nn * H1HEADS;
  float*    eexp  = ws + o; o += (size_t)ET * H1HEADS;
  float*    agg1  = ws + o; o += (size_t)Nn * HC1;
  float*    h1    = ws + o; o += (size_t)Nn * C;
  float*    pooled= ws + o; o += (size_t)G * C;
  float*    cnt   = ws + o; o += (size_t)G;
  float*    z     = ws + o; o += (size_t)G * C;
  // layer2 reuse (H1 dead after finalize1; agg1 dead after finalize2 source read)
  float* H2   = H1;                       // [N, C]
  float* agg2 = H1 + (size_t)Nn * C;      // [N, C]
  float* h2   = agg1;                     // [N, C]
  (void)ws_size; (void)n_in;

  const int TB = 256;                     // 8 wave32s per block

  // ================= Layer 1 =================
  {
    int waves = (Nn / 16) * (HC1 / 64);
    gemm_wmma_f32<<<ceil_div_i(waves, TB / 32), TB, 0, stream>>>(
        x, W1, H1, Nn, HC1, IN, nullptr, 0);

    attn_coef<<<ceil_div_i(Nn, TB / 32), TB, 0, stream>>>(
        H1, attS1, attD1, aS, aD, Nn, H1HEADS, C);

    long long nEH = (long long)Nn * H1HEADS;
    fill_u32<<<ceil_div_i(nEH, TB), TB, 0, stream>>>(emax, ENC_NEG_INF, nEH);
    fill_u32<<<ceil_div_i(nEH, TB), TB, 0, stream>>>((unsigned*)den, 0u, nEH);
    long long nAgg = (long long)Nn * HC1;
    fill_u32<<<ceil_div_i(nAgg, TB), TB, 0, stream>>>((unsigned*)agg1, 0u, nAgg);

    int eth = ET * H1HEADS;
    edge_max_k<<<ceil_div_i(eth, TB), TB, 0, stream>>>(
        srcI, dstI, aS, aD, emax, E, ET, H1HEADS);
    edge_expsum_k<<<ceil_div_i(eth, TB), TB, 0, stream>>>(
        srcI, dstI, aS, aD, emax, eexp, den, E, ET, H1HEADS);

    long long sc = (long long)ET * (HC1 / 4);
    edge_scatter_k<<<ceil_div_i(sc, TB), TB, 0, stream>>>(
        srcI, dstI, H1, eexp, den, agg1, E, ET, H1HEADS, C);

    finalize_elu<<<ceil_div_i((long long)Nn * C, TB), TB, 0, stream>>>(
        agg1, b1, h1, Nn, H1HEADS, C);
  }

  // ================= Layer 2 (heads = 1) =================
  {
    int waves = (Nn / 16) * (C / 64);
    gemm_wmma_f32<<<ceil_div_i(waves, TB / 32), TB, 0, stream>>>(
        h1, W2, H2, Nn, C, C, nullptr, 0);

    attn_coef<<<ceil_div_i(Nn, TB / 32), TB, 0, stream>>>(
        H2, attS2, attD2, aS, aD, Nn, 1, C);

    long long nEH = Nn;
    fill_u32<<<ceil_div_i(nEH, TB), TB, 0, stream>>>(emax, ENC_NEG_INF, nEH);
    fill_u32<<<ceil_div_i(nEH, TB), TB, 0, stream>>>((unsigned*)den, 0u, nEH);
    long long nAgg = (long long)Nn * C;
    fill_u32<<<ceil_div_i(nAgg, TB), TB, 0, stream>>>((unsigned*)agg2, 0u, nAgg);

    edge_max_k<<<ceil_div_i(ET, TB), TB, 0, stream>>>(
        srcI, dstI, aS, aD, emax, E, ET, 1);
    edge_expsum_k<<<ceil_div_i(ET, TB), TB, 0, stream>>>(
        srcI, dstI, aS, aD, emax, eexp, den, E, ET, 1);

    long long sc = (long long)ET * (C / 4);
    edge_scatter_k<<<ceil_div_i(sc, TB), TB, 0, stream>>>(
        srcI, dstI, H2, eexp, den, agg2, E, ET, 1, C);

    finalize_elu<<<ceil_div_i((long long)Nn * C, TB), TB, 0, stream>>>(
        agg2, b2, h2, Nn, 1, C);
  }

  // ================= Global mean pool =================
  {
    long long nP = (long long)G * C;
    fill_u32<<<ceil_div_i(nP, TB), TB, 0, stream>>>((unsigned*)pooled, 0u, nP);
    fill_u32<<<1, TB, 0, stream>>>((unsigned*)cnt, 0u, G);
    pool_add_k<<<ceil_div_i((long long)Nn * C, TB), TB, 0, stream>>>(h2, batch, pooled, Nn, C);
    count_k<<<ceil_div_i(Nn, TB), TB, 0, stream>>>(batch, cnt, Nn);
    pool_div_k<<<ceil_div_i(nP, TB), TB, 0, stream>>>(pooled, cnt, G, C);
  }

  // ================= MLP head =================
  {
    int waves = (G / 16) * (C / 64);          // 8 waves -> 1 block
    gemm_wmma_f32<<<ceil_div_i(waves, TB / 32), TB, 0, stream>>>(
        pooled, Wc1, z, G, C, C, bc1, 1);
    head_logits_k<<<1, ((G + 63) / 64) * 64, 0, stream>>>(z, Wc2, bc2, (float*)d_out, G, C);
  }
}